// EdgeAwareGINE_50208167690776
// MI455X (gfx1250) — compile-verified
//
#include <hip/hip_runtime.h>
#include <hip/hip_bf16.h>

typedef __attribute__((ext_vector_type(2))) float v2f;
typedef __attribute__((ext_vector_type(8))) float v8f;

#define HID  64
#define EDIM 32
#define BN_EPS 1e-5f

__device__ __forceinline__ v8f wmma4(v2f a, v2f b, v8f c) {
  // D = A(16x4 f32) * B(4x16 f32) + C(16x16 f32)
  return __builtin_amdgcn_wmma_f32_16x16x4_f32(
      /*neg_a=*/false, a, /*neg_b=*/false, b,
      /*c_mod=*/(short)0, c, /*reuse_a=*/false, /*reuse_b=*/false);
}

// -------------------------------------------------------------------------
// Edge kernel: fused  e = edge_attr @ We + be ; m = relu(x[src] + e) ;
//              atomic scatter-add of m into agg[dst].
// One wave handles 16 edges per tile. 32 WMMA (K=32, N=64) per tile.
// -------------------------------------------------------------------------
__global__ void __launch_bounds__(256) gine_edge_kernel(
    const float* __restrict__ x, const int* __restrict__ ei,
    const float* __restrict__ ea, const float* __restrict__ We,
    const float* __restrict__ be, float* __restrict__ agg, int E)
{
  const int lane = threadIdx.x & 31;
  const int half = lane >> 4;
  const int l16  = lane & 15;

  // Preload B fragments of We (32x64): Bw[kb8][nb], reused for every edge tile.
  v2f Bw[8][4];
#pragma unroll
  for (int kb8 = 0; kb8 < 8; ++kb8) {
#pragma unroll
    for (int nb = 0; nb < 4; ++nb) {
      const int k = kb8 * 4 + 2 * half;
      const int c = nb * 16 + l16;
      v2f b;
      b.x = We[k * HID + c];
      b.y = We[(k + 1) * HID + c];
      Bw[kb8][nb] = b;
    }
  }
  float bec[4];
#pragma unroll
  for (int nb = 0; nb < 4; ++nb) bec[nb] = be[nb * 16 + l16];

  const int wid = (blockIdx.x * blockDim.x + threadIdx.x) >> 5;
  const int nw  = (gridDim.x * blockDim.x) >> 5;
  const int nTiles = E >> 4;

  for (int t = wid; t < nTiles; t += nw) {
    const int e0 = t << 4;
    // lanes 0-15 hold src[e0+l16], lanes 16-31 hold dst[e0+l16]
    const int idxReg = ei[(size_t)half * (size_t)E + (size_t)(e0 + l16)];

    v8f acc[4] = {};
    const float* arow = ea + (size_t)(e0 + l16) * EDIM + 2 * half;
#pragma unroll
    for (int kb8 = 0; kb8 < 8; ++kb8) {
      const v2f a = *(const v2f*)(arow + kb8 * 4);
#pragma unroll
      for (int nb = 0; nb < 4; ++nb)
        acc[nb] = wmma4(a, Bw[kb8][nb], acc[nb]);
    }

#pragma unroll
    for (int v = 0; v < 8; ++v) {
      const int r = v + 8 * half;                 // D-matrix row for this lane
      const int s = __shfl(idxReg, r, 32);        // src node of edge e0+r
      const int d = __shfl(idxReg, r + 16, 32);   // dst node of edge e0+r
      const float* xr = x   + (size_t)s * HID;
      float*       ar = agg + (size_t)d * HID;
#pragma unroll
      for (int nb = 0; nb < 4; ++nb) {
        const int c = nb * 16 + l16;
        float m = acc[nb][v] + bec[nb] + xr[c];
        m = fmaxf(m, 0.0f);
        atomicAdd(ar + c, m);                     // global_atomic_add_f32, no-return
      }
    }
  }
}

// -------------------------------------------------------------------------
// MLP kernel:  h = relu((x+agg) @ W1 + b1) @ W2 + b2 ; accumulate BN stats.
// Block = 4 waves; W1/W2 staged in LDS; one 16-row tile per wave iteration.
// -------------------------------------------------------------------------
__global__ void __launch_bounds__(128) gine_mlp_kernel(
    const float* __restrict__ x, const float* __restrict__ agg,
    const float* __restrict__ W1, const float* __restrict__ b1,
    const float* __restrict__ W2, const float* __restrict__ b2,
    float* __restrict__ h, float* __restrict__ stats, int N)
{
  __shared__ float W1s[HID * HID];
  __shared__ float W2s[HID * HID];
  __shared__ float hT[4][16 * 68];   // stride 68 -> conflict-free b64 A-reads
  __shared__ float statS[128];

  const int tid  = threadIdx.x;
  const int wave = tid >> 5, lane = tid & 31;
  const int half = lane >> 4, l16 = lane & 15;

  for (int i = tid * 4; i < HID * HID; i += 128 * 4) {
    *(float4*)(W1s + i) = *(const float4*)(W1 + i);
    *(float4*)(W2s + i) = *(const float4*)(W2 + i);
  }
  statS[tid] = 0.0f;
  __syncthreads();

  float b1c[4], b2c[4];
#pragma unroll
  for (int nb = 0; nb < 4; ++nb) {
    b1c[nb] = b1[nb * 16 + l16];
    b2c[nb] = b2[nb * 16 + l16];
  }
  float lsum[4] = {0.f, 0.f, 0.f, 0.f};
  float lsq [4] = {0.f, 0.f, 0.f, 0.f};

  float* ht = hT[wave];
  const int wid = blockIdx.x * 4 + wave;
  const int nw  = gridDim.x * 4;
  const int nTiles = N >> 4;

  for (int t = wid; t < nTiles; t += nw) {
    const int r0 = t << 4;
    // stage x+agg tile into LDS
#pragma unroll
    for (int q = 0; q < 8; ++q) {
      const int flat = q * 32 + lane;
      const int row = flat >> 4, c4 = (flat & 15) << 2;
      const float4 xv = *(const float4*)(x   + (size_t)(r0 + row) * HID + c4);
      const float4 av = *(const float4*)(agg + (size_t)(r0 + row) * HID + c4);
      float4 s; s.x = xv.x + av.x; s.y = xv.y + av.y; s.z = xv.z + av.z; s.w = xv.w + av.w;
      *(float4*)(ht + row * 68 + c4) = s;
    }
    // GEMM1: (x+agg) @ W1
    v8f acc[4] = {};
#pragma unroll
    for (int kb = 0; kb < 16; ++kb) {
      const v2f a = *(const v2f*)(ht + l16 * 68 + kb * 4 + 2 * half);
      const int k = kb * 4 + 2 * half;
#pragma unroll
      for (int nb = 0; nb < 4; ++nb) {
        const int c = nb * 16 + l16;
        v2f b; b.x = W1s[k * HID + c]; b.y = W1s[(k + 1) * HID + c];
        acc[nb] = wmma4(a, b, acc[nb]);
      }
    }
    // relu(.. + b1) written back to LDS tile (same wave -> DScnt-ordered)
#pragma unroll
    for (int nb = 0; nb < 4; ++nb)
#pragma unroll
      for (int v = 0; v < 8; ++v)
        ht[(v + 8 * half) * 68 + nb * 16 + l16] = fmaxf(acc[nb][v] + b1c[nb], 0.0f);
    // GEMM2: t @ W2
    v8f acc2[4] = {};
#pragma unroll
    for (int kb = 0; kb < 16; ++kb) {
      const v2f a = *(const v2f*)(ht + l16 * 68 + kb * 4 + 2 * half);
      const int k = kb * 4 + 2 * half;
#pragma unroll
      for (int nb = 0; nb < 4; ++nb) {
        const int c = nb * 16 + l16;
        v2f b; b.x = W2s[k * HID + c]; b.y = W2s[(k + 1) * HID + c];
        acc2[nb] = wmma4(a, b, acc2[nb]);
      }
    }
    // epilogue: write h, accumulate per-column stats in registers
#pragma unroll
    for (int nb = 0; nb < 4; ++nb) {
      const int c = nb * 16 + l16;
      float s = 0.f, s2 = 0.f;
#pragma unroll
      for (int v = 0; v < 8; ++v) {
        const float hv = acc2[nb][v] + b2c[nb];
        h[(size_t)(r0 + v + 8 * half) * HID + c] = hv;
        s += hv; s2 += hv * hv;
      }
      lsum[nb] += s; lsq[nb] += s2;
    }
  }

#pragma unroll
  for (int nb = 0; nb < 4; ++nb) {
    atomicAdd(&statS[nb * 16 + l16],      lsum[nb]);   // ds_add_f32
    atomicAdd(&statS[64 + nb * 16 + l16], lsq[nb]);
  }
  __syncthreads();
  atomicAdd(&stats[tid], statS[tid]);   // one global atomic per column per block
}

// -------------------------------------------------------------------------
// BN finalize: fold mean/var/gamma/beta into per-column scale/shift.
// -------------------------------------------------------------------------
__global__ void gine_bn_finalize(float* __restrict__ stats,
                                 const float* __restrict__ gamma,
                                 const float* __restrict__ beta, float invN)
{
  const int c = threadIdx.x;   // 64 threads
  const float mean = stats[c] * invN;
  const float var  = stats[64 + c] * invN - mean * mean;
  const float sc   = gamma[c] * rsqrtf(var + BN_EPS);
  stats[128 + c] = sc;
  stats[192 + c] = beta[c] - mean * sc;
}

// -------------------------------------------------------------------------
// BN apply + relu -> next x (float4 elementwise)
// -------------------------------------------------------------------------
__global__ void __launch_bounds__(256) gine_bn_apply(
    const float* __restrict__ h, const float* __restrict__ stats,
    float* __restrict__ xo, int total4)
{
  const int i = blockIdx.x * blockDim.x + threadIdx.x;
  if (i >= total4) return;
  const int c4 = (i & 15) << 2;
  const float4 hv = *(const float4*)(h + (size_t)i * 4);
  float4 o;
  o.x = fmaxf(hv.x * stats[128 + c4 + 0] + stats[192 + c4 + 0], 0.f);
  o.y = fmaxf(hv.y * stats[128 + c4 + 1] + stats[192 + c4 + 1], 0.f);
  o.z = fmaxf(hv.z * stats[128 + c4 + 2] + stats[192 + c4 + 2], 0.f);
  o.w = fmaxf(hv.w * stats[128 + c4 + 3] + stats[192 + c4 + 3], 0.f);
  *(float4*)(xo + (size_t)i * 4) = o;
}

// -------------------------------------------------------------------------
// Output kernel: out = x @ Wout + bout
// -------------------------------------------------------------------------
__global__ void __launch_bounds__(128) gine_out_kernel(
    const float* __restrict__ x, const float* __restrict__ Wo,
    const float* __restrict__ bo, float* __restrict__ out, int N)
{
  __shared__ float Ws[HID * HID];
  __shared__ float hT[4][16 * 68];

  const int tid  = threadIdx.x;
  const int wave = tid >> 5, lane = tid & 31;
  const int half = lane >> 4, l16 = lane & 15;

  for (int i = tid * 4; i < HID * HID; i += 128 * 4)
    *(float4*)(Ws + i) = *(const float4*)(Wo + i);
  __syncthreads();

  float boc[4];
#pragma unroll
  for (int nb = 0; nb < 4; ++nb) boc[nb] = bo[nb * 16 + l16];

  float* ht = hT[wave];
  const int wid = blockIdx.x * 4 + wave;
  const int nw  = gridDim.x * 4;
  const int nTiles = N >> 4;

  for (int t = wid; t < nTiles; t += nw) {
    const int r0 = t << 4;
#pragma unroll
    for (int q = 0; q < 8; ++q) {
      const int flat = q * 32 + lane;
      const int row = flat >> 4, c4 = (flat & 15) << 2;
      *(float4*)(ht + row * 68 + c4) =
          *(const float4*)(x + (size_t)(r0 + row) * HID + c4);
    }
    v8f acc[4] = {};
#pragma unroll
    for (int kb = 0; kb < 16; ++kb) {
      const v2f a = *(const v2f*)(ht + l16 * 68 + kb * 4 + 2 * half);
      const int k = kb * 4 + 2 * half;
#pragma unroll
      for (int nb = 0; nb < 4; ++nb) {
        const int c = nb * 16 + l16;
        v2f b; b.x = Ws[k * HID + c]; b.y = Ws[(k + 1) * HID + c];
        acc[nb] = wmma4(a, b, acc[nb]);
      }
    }
#pragma unroll
    for (int nb = 0; nb < 4; ++nb) {
      const int c = nb * 16 + l16;
#pragma unroll
      for (int v = 0; v < 8; ++v)
        out[(size_t)(r0 + v + 8 * half) * HID + c] = acc[nb][v] + boc[nb];
    }
  }
}

// -------------------------------------------------------------------------
extern "C" void kernel_launch(void* const* d_in, const int* in_sizes, int n_in,
                              void* d_out, int out_size, void* d_ws, size_t ws_size,
                              hipStream_t stream) {
  (void)n_in; (void)out_size; (void)ws_size;
  const float* x     = (const float*)d_in[0];
  const int*   ei    = (const int*)  d_in[1];
  const float* ea    = (const float*)d_in[2];
  const float* We    = (const float*)d_in[3];
  const float* be    = (const float*)d_in[4];
  const float* W1    = (const float*)d_in[5];
  const float* b1    = (const float*)d_in[6];
  const float* W2    = (const float*)d_in[7];
  const float* b2    = (const float*)d_in[8];
  const float* gamma = (const float*)d_in[9];
  const float* beta  = (const float*)d_in[10];
  const float* Wout  = (const float*)d_in[11];
  const float* bout  = (const float*)d_in[12];

  const int N = in_sizes[0] / HID;   // 100000
  const int E = in_sizes[1] / 2;     // 1600000

  float* ws    = (float*)d_ws;
  float* xbuf  = ws;                              // [N,64]
  float* agg   = xbuf + (size_t)N * HID;          // [N,64]
  float* hbuf  = agg  + (size_t)N * HID;          // [N,64]
  float* stats = hbuf + (size_t)N * HID;          // [256]

  const float invN = 1.0f / (float)N;
  const float* xin = x;

  for (int l = 0; l < 3; ++l) {
    hipMemsetAsync(agg, 0, (size_t)N * HID * sizeof(float), stream);
    hipMemsetAsync(stats, 0, 256 * sizeof(float), stream);

    gine_edge_kernel<<<1280, 256, 0, stream>>>(
        xin, ei, ea, We + (size_t)l * EDIM * HID, be + (size_t)l * HID, agg, E);

    gine_mlp_kernel<<<390, 128, 0, stream>>>(
        xin, agg,
        W1 + (size_t)l * HID * HID, b1 + (size_t)l * HID,
        W2 + (size_t)l * HID * HID, b2 + (size_t)l * HID,
        hbuf, stats, N);

    gine_bn_finalize<<<1, 64, 0, stream>>>(
        stats, gamma + (size_t)l * HID, beta + (size_t)l * HID, invN);

    gine_bn_apply<<<(N * 16 + 255) / 256, 256, 0, stream>>>(
        hbuf, stats, xbuf, N * 16);

    xin = xbuf;
  }

  gine_out_kernel<<<390, 128, 0, stream>>>(xin, Wout, bout, (float*)d_out, N);
}